// Sampler_27616639713591
// MI455X (gfx1250) — compile-verified
//
#include <hip/hip_runtime.h>
#include <hip/hip_bf16.h>
#include <math.h>

#define V_DIM 128000
#define H_DIM 2048
#define B_DIM 32
#define S_DIM 1024
#define SOFTCAP 30.0f

typedef __attribute__((ext_vector_type(2))) float v2f;
typedef __attribute__((ext_vector_type(8))) float v8f;

// ---------------------------------------------------------------------------
// Kernel 1: logits[32, 128000] = hs @ emb^T, softcap + temperature, via
// V_WMMA_F32_16X16X4_F32. One wave = 16 vocab columns x all 32 rows
// (two 16x16 f32 accumulators sharing the B fragment). Memory-bound on the
// 1.05 GB fp32 embedding stream -> fp32 WMMA is the right precision & fast
// enough by ~10x.
// ---------------------------------------------------------------------------
__global__ __launch_bounds__(256) void logits_wmma_kernel(
    const float* __restrict__ emb,      // [V, H]
    const float* __restrict__ hidden,   // [B, S, H]
    const int*   __restrict__ pos_p,    // [1]
    const float* __restrict__ temps,    // [B]
    float*       __restrict__ out)      // d_out: [32 ids][32*128000 logits]
{
    const int pos  = pos_p[0];
    const int lane = threadIdx.x & 31;
    const int wave = threadIdx.x >> 5;
    const int lo   = lane & 15;          // M (for A) / N (for B) within tile
    const int hi   = lane >> 4;          // selects K pair {0,1} vs {2,3}

    const int n_base = blockIdx.x * 128 + wave * 16;

    // B fragment source: B[k][n] = emb[n_base+n][k]; lane lo holds row n_base+lo,
    // VGPR j = K = k0 + 2*hi + j  -> contiguous 8B load at (row*H + k0 + 2*hi).
    const float* bptr  = emb + (size_t)(n_base + lo) * H_DIM + 2 * hi;
    // A fragments: A[m][k] = hs[m][k] = hidden[m, pos, k]; same per-lane pattern.
    const float* a0ptr = hidden + ((size_t)lo        * S_DIM + pos) * H_DIM + 2 * hi;
    const float* a1ptr = hidden + ((size_t)(lo + 16) * S_DIM + pos) * H_DIM + 2 * hi;

    v8f acc0 = {};   // rows 0..15
    v8f acc1 = {};   // rows 16..31

    for (int kk = 0; kk < H_DIM; kk += 64) {
        // keep the embedding stream ahead of the WMMA pipe
        __builtin_prefetch(bptr + kk + 512, 0, 1);
#pragma unroll
        for (int k4 = 0; k4 < 64; k4 += 4) {
            const int k0 = kk + k4;
            v2f a0 = *(const v2f*)(a0ptr + k0);
            v2f a1 = *(const v2f*)(a1ptr + k0);
            v2f b  = *(const v2f*)(bptr  + k0);
            acc0 = __builtin_amdgcn_wmma_f32_16x16x4_f32(
                false, a0, false, b, (short)0, acc0, false, false);
            acc1 = __builtin_amdgcn_wmma_f32_16x16x4_f32(
                false, a1, false, b, (short)0, acc1, false, false);
        }
    }

    // Epilogue: softcap tanh + temperature, store per C/D layout
    // (VGPR r: lanes 0-15 -> M=r, lanes 16-31 -> M=r+8, N = lo).
    float* logits = out + B_DIM;   // ids occupy out[0..31]
    const float inv_cap = 1.0f / SOFTCAP;
#pragma unroll
    for (int r = 0; r < 8; ++r) {
        int   m0 = r + 8 * hi;
        float x0 = tanhf(acc0[r] * inv_cap) * SOFTCAP / temps[m0];
        logits[(size_t)m0 * V_DIM + n_base + lo] = x0;

        int   m1 = m0 + 16;
        float x1 = tanhf(acc1[r] * inv_cap) * SOFTCAP / temps[m1];
        logits[(size_t)m1 * V_DIM + n_base + lo] = x1;
    }
}

// ---------------------------------------------------------------------------
// Kernel 2: per-row softmax stats (max, sum of exp). Logits are L2-resident.
// ---------------------------------------------------------------------------
__global__ __launch_bounds__(256) void row_stats_kernel(
    const float* __restrict__ logits, float* __restrict__ ws)
{
    __shared__ float red[256];
    const int b = blockIdx.x;
    const int t = threadIdx.x;
    const float* row = logits + (size_t)b * V_DIM;

    float mx = -INFINITY;
    for (int i = t; i < V_DIM; i += 256) mx = fmaxf(mx, row[i]);
    red[t] = mx; __syncthreads();
    for (int s = 128; s > 0; s >>= 1) {
        if (t < s) red[t] = fmaxf(red[t], red[t + s]);
        __syncthreads();
    }
    const float m = red[0];
    __syncthreads();

    float sum = 0.0f;
    for (int i = t; i < V_DIM; i += 256) sum += expf(row[i] - m);
    red[t] = sum; __syncthreads();
    for (int s = 128; s > 0; s >>= 1) {
        if (t < s) red[t] += red[t + s];
        __syncthreads();
    }
    if (t == 0) { ws[b] = m; ws[B_DIM + b] = red[0]; }
}

// ---------------------------------------------------------------------------
// Kernel 3: exact top-64 selection per row (top_ks < 64 => support <= 63
// tokens), then top-p / top-k mask, renormalize, inverse-CDF sample with a
// deterministic per-row hash.
// ---------------------------------------------------------------------------
__global__ __launch_bounds__(256) void topk_sample_kernel(
    const float* __restrict__ logits, const float* __restrict__ ws,
    const float* __restrict__ top_ps, const int* __restrict__ top_ks,
    float* __restrict__ out_ids)
{
    __shared__ float rv[256];
    __shared__ int   ri[256];
    __shared__ float sel_v[64];
    __shared__ int   sel_i[64];
    __shared__ float bc_v;
    __shared__ int   bc_i;

    const int b = blockIdx.x;
    const int t = threadIdx.x;
    const float* row = logits + (size_t)b * V_DIM;

    float prev_v = INFINITY;
    int   prev_i = -1;

    for (int it = 0; it < 64; ++it) {
        float best_v = -INFINITY;
        int   best_i = 0x7FFFFFFF;
        for (int i = t; i < V_DIM; i += 256) {
            float v = row[i];
            // enumerate strictly after (prev_v, prev_i) in descending
            // (value, then ascending index) order
            bool elig = (v < prev_v) || (v == prev_v && i > prev_i);
            if (elig && (v > best_v || (v == best_v && i < best_i))) {
                best_v = v; best_i = i;
            }
        }
        rv[t] = best_v; ri[t] = best_i; __syncthreads();
        for (int s = 128; s > 0; s >>= 1) {
            if (t < s) {
                if (rv[t + s] > rv[t] ||
                    (rv[t + s] == rv[t] && ri[t + s] < ri[t])) {
                    rv[t] = rv[t + s]; ri[t] = ri[t + s];
                }
            }
            __syncthreads();
        }
        if (t == 0) { sel_v[it] = rv[0]; sel_i[it] = ri[0]; bc_v = rv[0]; bc_i = ri[0]; }
        __syncthreads();
        prev_v = bc_v; prev_i = bc_i;
        __syncthreads();
    }

    if (t == 0) {
        const float m  = ws[b];
        const float Z  = ws[B_DIM + b];
        const float tp = top_ps[b];
        int k = top_ks[b];
        if (k < 1)  k = 1;
        if (k > 64) k = 64;

        float kept[64];
        float cum_excl = 0.0f, kept_sum = 0.0f;
        for (int i = 0; i < 64; ++i) {
            float pi = expf(sel_v[i] - m) / Z;
            float keep = pi;
            if (cum_excl > tp) keep = 0.0f;   // (cumsum - p_i) > top_p
            if (i >= k)        keep = 0.0f;   // rank >= top_k
            cum_excl += pi;
            kept[i] = keep;
            kept_sum += keep;
        }

        // deterministic per-row uniform (stand-in for jax key(42) stream)
        unsigned h = (unsigned)b * 2654435761u ^ 0x9E3779B9u;
        h ^= h >> 16; h *= 0x85EBCA6Bu; h ^= h >> 13; h *= 0xC2B2AE35u; h ^= h >> 16;
        float u = (float)(h & 0xFFFFFFu) * (1.0f / 16777216.0f) * kept_sum;

        int pick = sel_i[0];
        float acc = 0.0f;
        for (int i = 0; i < 64; ++i) {
            acc += kept[i];
            if (kept[i] > 0.0f && acc >= u) { pick = sel_i[i]; break; }
        }
        out_ids[b] = (float)pick;
    }
}

// ---------------------------------------------------------------------------
extern "C" void kernel_launch(void* const* d_in, const int* in_sizes, int n_in,
                              void* d_out, int out_size, void* d_ws, size_t ws_size,
                              hipStream_t stream) {
    const float* emb    = (const float*)d_in[0];   // [V, H]
    const float* hidden = (const float*)d_in[1];   // [B, S, H]
    const int*   pos    = (const int*)  d_in[2];   // [1]
    const float* top_ps = (const float*)d_in[3];   // [B]
    const int*   top_ks = (const int*)  d_in[4];   // [B]
    const float* temps  = (const float*)d_in[5];   // [B]
    float* out = (float*)d_out;                    // [32 ids][32*128000 logits]
    float* ws  = (float*)d_ws;                     // [32 max][32 sumexp]

    logits_wmma_kernel<<<V_DIM / 128, 256, 0, stream>>>(emb, hidden, pos, temps, out);
    row_stats_kernel  <<<B_DIM, 256, 0, stream>>>(out + B_DIM, ws);
    topk_sample_kernel<<<B_DIM, 256, 0, stream>>>(out + B_DIM, ws, top_ps, top_ks, out);
}